// VectorQuantizer_4389456576698
// MI455X (gfx1250) — compile-verified
//
#include <hip/hip_runtime.h>
#include <math.h>

// ---------------------------------------------------------------------------
// VQ (VQ-VAE vector quantizer) forward for MI455X / gfx1250, wave32 + WMMA.
//
// rows = 65536 (8192 x 8), dim = 64, codebook = 1024 x 64 (256 KB).
// argmin_n ||z - e_n||^2  ==  argmin_n ( ||e_n||^2 - 2 z.e_n ).
//
// Main kernel: 512 threads (16 waves), 256 z-rows per block.
//  - codebook staged into LDS once per block via global_load_async_to_lds_b128
//    (ASYNCcnt), rows padded to 68 floats for conflict-free ds_load_b64
//  - each wave: 16 rows x all 1024 columns; per column-tile-pair, two
//    interleaved 16-deep V_WMMA_F32_16X16X4_F32 chains (K=64)
//  - per-lane argmin -> intra-wave shfl_xor butterfly -> row indices
//  - z_q gathered from LDS codebook; loss partials via LDS tree + atomicAdd
// ---------------------------------------------------------------------------

#define ROWS   65536
#define NE     1024
#define DIM    64
#define EPAD   68                       // padded row stride (floats)
#define ROWS_PER_BLOCK 256
#define THREADS 512

typedef float v2f __attribute__((ext_vector_type(2)));
typedef float v4f __attribute__((ext_vector_type(4)));
typedef float v8f __attribute__((ext_vector_type(8)));

// ---- output offsets (floats) in d_out, reference tuple order ----
#define OFF_LOSS   ((size_t)0)
#define OFF_LOSS1  ((size_t)1)
#define OFF_LOSS2  ((size_t)2)
#define OFF_ZQ     ((size_t)3)
#define OFF_ZOUT   (OFF_ZQ + (size_t)ROWS * DIM)          // 3 + 4194304
#define OFF_PERP   (OFF_ZOUT + (size_t)ROWS * DIM)        // 8388611
#define OFF_ONEHOT (OFF_PERP + 1)                         // 8388612
#define OFF_IDX    (OFF_ONEHOT + (size_t)ROWS * NE)       // 75497476

// ---- workspace layout (bytes) ----
#define WS_COUNTS  0          // u32[1024]
#define WS_SUMSQ   4096       // float
#define WS_IDX     8192       // int[65536]

// ---- dynamic LDS layout (bytes); no static LDS in main kernel, so the
// dynamic region starts at LDS offset 0 (used by the async-load VDST) ----
#define LDS_EMB_BYTES   ((size_t)NE * EPAD * 4)           // 278528
#define LDS_ROWIDX_OFF  LDS_EMB_BYTES
#define LDS_RED_OFF     (LDS_EMB_BYTES + ROWS_PER_BLOCK * 4)
#define LDS_TOTAL       (LDS_RED_OFF + THREADS * 4)       // 281600 < 320KB

__global__ __launch_bounds__(1024) void vq_init_kernel(unsigned int* counts,
                                                       float* sumSq) {
  int t = threadIdx.x;
  if (t < NE) counts[t] = 0u;
  if (t == 0) *sumSq = 0.0f;
}

__global__ __launch_bounds__(THREADS) void vq_main_kernel(
    const float* __restrict__ z, const float* __restrict__ emb,
    float* __restrict__ zq_out, float* __restrict__ zout2,
    float* __restrict__ idx_out_f, int* __restrict__ idx_ws,
    unsigned int* __restrict__ counts, float* __restrict__ sumSq) {
  extern __shared__ char smemRaw[];
  float* sEmb    = (float*)smemRaw;                 // LDS offset 0
  int*   sRowIdx = (int*)(smemRaw + LDS_ROWIDX_OFF);
  float* sRed    = (float*)(smemRaw + LDS_RED_OFF);

  const int tid     = threadIdx.x;
  const int wave    = tid >> 5;
  const int lane    = tid & 31;
  const int laneLow = lane & 15;
  const int laneHi  = lane >> 4;          // 0: K%4 in {0,1}; 1: K%4 in {2,3}
  const int baseRow = blockIdx.x * ROWS_PER_BLOCK;

  // ---- stage codebook (1024 x 64 f32) into LDS with padded rows via the
  // gfx1250 async DMA path. chunk c = row*4 + part (16B each); LDS dest is a
  // raw byte offset (dynamic LDS region starts at 0).
#pragma unroll
  for (int it = 0; it < 8; ++it) {
    const int c    = it * THREADS + tid;         // 0..4095
    const int row  = c >> 2;
    const int part = c & 3;
    unsigned dstOff = (unsigned)(row * (EPAD * 4) + part * 16);
    unsigned long long gaddr =
        (unsigned long long)((const char*)emb + row * 256 + part * 16);
    asm volatile("global_load_async_to_lds_b128 %0, %1, off"
                 :: "v"(dstOff), "v"(gaddr) : "memory");
  }

  // ---- A fragments for this wave's 16 rows (overlaps with async staging).
  // Lane l holds A row l%16, K = 4*ks + 2*laneHi + {0,1}.
  const float* zrow =
      z + (size_t)(baseRow + wave * 16 + laneLow) * DIM + 2 * laneHi;
  v2f a[16];
#pragma unroll
  for (int ks = 0; ks < 16; ++ks)
    a[ks] = *(const v2f*)(zrow + 4 * ks);

  asm volatile("s_wait_asynccnt 0x0" ::: "memory");
  __syncthreads();

  float best[8];
  int   bestIdx[8];
#pragma unroll
  for (int r = 0; r < 8; ++r) { best[r] = 3.4e38f; bestIdx[r] = 0x7fffffff; }

  // ---- 64 column tiles, processed as 32 pairs with two interleaved
  // K-chained WMMA accumulators for matrix-pipe ILP.
  for (int tp = 0; tp < 32; ++tp) {
    const int nBase0 = tp * 32;
    const float* e0 = sEmb + (nBase0 + laneLow) * EPAD + 2 * laneHi;
    const float* e1 = e0 + 16 * EPAD;
    v2f b0[16], b1[16];
#pragma unroll
    for (int ks = 0; ks < 16; ++ks) {       // batched ds_load_b64, one wait
      b0[ks] = *(const v2f*)(e0 + 4 * ks);
      b1[ks] = *(const v2f*)(e1 + 4 * ks);
    }
    v8f acc0 = {}, acc1 = {};
    float en0 = 0.0f, en1 = 0.0f;
#pragma unroll
    for (int ks = 0; ks < 16; ++ks) {
      en0 += b0[ks].x * b0[ks].x + b0[ks].y * b0[ks].y;
      en1 += b1[ks].x * b1[ks].x + b1[ks].y * b1[ks].y;
      acc0 = __builtin_amdgcn_wmma_f32_16x16x4_f32(
          false, a[ks], false, b0[ks], (short)0, acc0, false, false);
      acc1 = __builtin_amdgcn_wmma_f32_16x16x4_f32(
          false, a[ks], false, b1[ks], (short)0, acc1, false, false);
    }
    // lanes l and l^16 share column n and jointly hold the full emb row
    en0 += __shfl_xor(en0, 16, 32);
    en1 += __shfl_xor(en1, 16, 32);
    const int n0 = nBase0 + laneLow;
    const int n1 = n0 + 16;
#pragma unroll
    for (int r = 0; r < 8; ++r) {
      float d0 = en0 - 2.0f * acc0[r];      // ||e||^2 - 2 z.e
      if (d0 < best[r]) { best[r] = d0; bestIdx[r] = n0; }
      float d1 = en1 - 2.0f * acc1[r];
      if (d1 < best[r]) { best[r] = d1; bestIdx[r] = n1; }
    }
  }

  // ---- intra-wave argmin butterfly across the 16 lanes of each half
  // (columns are distributed mod 16 across lanes; ties -> lowest index).
#pragma unroll
  for (int r = 0; r < 8; ++r) {
    float v = best[r];
    int   i = bestIdx[r];
#pragma unroll
    for (int m = 8; m >= 1; m >>= 1) {
      float vo = __shfl_xor(v, m, 32);
      int   io = __shfl_xor(i, m, 32);
      if (vo < v || (vo == v && io < i)) { v = vo; i = io; }
    }
    best[r] = v;
    bestIdx[r] = i;
  }

  // lanes 0 and 16 hold final results for rows {0..7} / {8..15} of the wave
  if (laneLow == 0) {
#pragma unroll
    for (int r = 0; r < 8; ++r) {
      const int localRow = wave * 16 + 8 * laneHi + r;
      const int row = baseRow + localRow;
      const int bi = bestIdx[r];
      sRowIdx[localRow] = bi;
      idx_ws[row]       = bi;
      idx_out_f[row]    = (float)bi;
    }
  }
  __syncthreads();

  if (tid < ROWS_PER_BLOCK) atomicAdd(&counts[sRowIdx[tid]], 1u);

  // ---- gather z_q from the LDS codebook, write both copies, loss partial
  float localSq = 0.0f;
#pragma unroll
  for (int e = tid; e < ROWS_PER_BLOCK * DIM; e += THREADS) {
    const int m = e >> 6;
    const int j = e & 63;
    const int ci = sRowIdx[m];
    const size_t off = (size_t)(baseRow + m) * DIM + j;
    float q  = sEmb[ci * EPAD + j];
    float zv = z[off];
    float d  = q - zv;
    localSq += d * d;
    zq_out[off] = q;
    zout2[off]  = q;
  }
  sRed[tid] = localSq;
  __syncthreads();
  for (int s = THREADS / 2; s > 0; s >>= 1) {
    if (tid < s) sRed[tid] += sRed[tid + s];
    __syncthreads();
  }
  if (tid == 0) atomicAdd(sumSq, sRed[0]);
}

// 256 MB zero-fill; > L2 (192MB), so use non-temporal 128-bit stores.
__global__ __launch_bounds__(256) void vq_onehot_zero_kernel(v4f* p,
                                                             size_t n4) {
  size_t i      = (size_t)blockIdx.x * blockDim.x + threadIdx.x;
  size_t stride = (size_t)gridDim.x * blockDim.x;
  v4f z4 = {0.0f, 0.0f, 0.0f, 0.0f};
  for (; i < n4; i += stride) __builtin_nontemporal_store(z4, &p[i]);
}

__global__ __launch_bounds__(256) void vq_onehot_scatter_kernel(
    float* __restrict__ onehot, const int* __restrict__ idx) {
  int row = blockIdx.x * blockDim.x + threadIdx.x;
  if (row < ROWS) onehot[(size_t)row * NE + idx[row]] = 1.0f;
}

__global__ __launch_bounds__(1024) void vq_finalize_kernel(
    const unsigned int* __restrict__ counts, const float* __restrict__ sumSq,
    float* __restrict__ out_scalars, float* __restrict__ out_perp) {
  __shared__ float sred[1024];
  int t = threadIdx.x;
  float p = (float)counts[t] * (1.0f / (float)ROWS);
  sred[t] = p * logf(p + 1e-10f);
  __syncthreads();
  for (int s = 512; s > 0; s >>= 1) {
    if (t < s) sred[t] += sred[t + s];
    __syncthreads();
  }
  if (t == 0) {
    float l1 = *sumSq * (1.0f / (float)((size_t)ROWS * DIM));
    // z_q_st == z_q numerically -> loss1 == loss2; loss = 1.0*l1 + 0.25*l2
    out_scalars[OFF_LOSS]  = 1.25f * l1;
    out_scalars[OFF_LOSS1] = l1;
    out_scalars[OFF_LOSS2] = l1;
    *out_perp = expf(-sred[0]);
  }
}

extern "C" void kernel_launch(void* const* d_in, const int* in_sizes, int n_in,
                              void* d_out, int out_size, void* d_ws,
                              size_t ws_size, hipStream_t stream) {
  const float* z   = (const float*)d_in[0];   // 8192*512
  const float* emb = (const float*)d_in[1];   // 1024*64
  float* out = (float*)d_out;

  unsigned int* counts = (unsigned int*)((char*)d_ws + WS_COUNTS);
  float*        sumSq  = (float*)((char*)d_ws + WS_SUMSQ);
  int*          idxWs  = (int*)((char*)d_ws + WS_IDX);

  // 0) zero accumulators (workspace is poisoned, must init every call)
  vq_init_kernel<<<1, 1024, 0, stream>>>(counts, sumSq);

  // 1) fused WMMA distances + argmin + gather + loss partials
  vq_main_kernel<<<ROWS / ROWS_PER_BLOCK, THREADS, LDS_TOTAL, stream>>>(
      z, emb, out + OFF_ZQ, out + OFF_ZOUT, out + OFF_IDX, idxWs, counts,
      sumSq);

  // 2) one-hot zero fill (NT 128-bit stores): 67108864 floats = 16777216 v4f
  const size_t n4 = (size_t)ROWS * NE / 4;
  vq_onehot_zero_kernel<<<4096, 256, 0, stream>>>(
      (v4f*)(out + OFF_ONEHOT), n4);

  // 3) scatter the ones
  vq_onehot_scatter_kernel<<<ROWS / 256, 256, 0, stream>>>(out + OFF_ONEHOT,
                                                           idxWs);

  // 4) scalars: loss, loss1, loss2, perplexity
  vq_finalize_kernel<<<1, 1024, 0, stream>>>(counts, sumSq, out,
                                             out + OFF_PERP);
}